// HingeLoss_1236950581440
// MI455X (gfx1250) — compile-verified
//
#include <hip/hip_runtime.h>
#include <stdint.h>

typedef __attribute__((ext_vector_type(2))) float v2f;
typedef __attribute__((ext_vector_type(4))) float v4f;
typedef __attribute__((ext_vector_type(8))) float v8f;

#define D 128
#define DELTA_V 1.0f

// ---- order-preserving float <-> uint32 key (for atomicMax on floats) ----
__device__ __forceinline__ unsigned f2key(float x) {
    unsigned b = __float_as_uint(x);
    return (b & 0x80000000u) ? ~b : (b | 0x80000000u);
}
__device__ __forceinline__ float key2f(unsigned k) {
    unsigned b = (k & 0x80000000u) ? (k ^ 0x80000000u) : ~k;
    return __uint_as_float(b);
}

// ---- zero accumulators (d_out / d_ws are poisoned by the harness) ----
__global__ void zero_kernel(float* __restrict__ p_sum, unsigned* __restrict__ p_cnt,
                            unsigned* __restrict__ n_key, float* __restrict__ out, int n) {
    int i = blockIdx.x * blockDim.x + threadIdx.x;
    if (i < n) { p_sum[i] = 0.0f; p_cnt[i] = 0u; n_key[i] = 0u; }
    if (i == 0) out[0] = 0.0f;
}

// ---- gather emb[nodes] into an L2-resident 131072x128 table + row norms ----
// one wave per row: 32 lanes x float4 = 128 floats.
// emb reads are single-touch from a 256MB table -> non-temporal, so the 192MB
// L2 keeps capacity for the 64MB `g` table that both edge passes re-read.
__global__ void gather_norm_kernel(const float* __restrict__ emb, const int* __restrict__ nodes,
                                   float* __restrict__ g, float* __restrict__ norms, int n_nodes) {
    int wave = (blockIdx.x * blockDim.x + threadIdx.x) >> 5;
    int lane = threadIdx.x & 31;
    if (wave >= n_nodes) return;
    int row = __builtin_nontemporal_load(nodes + wave);
    const v4f* src = (const v4f*)(emb + (size_t)row * D);
    v4f*       dst = (v4f*)(g + (size_t)wave * D);
    v4f v = __builtin_nontemporal_load(src + lane);
    dst[lane] = v;                                   // RT store: keep g hot in L2
    float s = v[0] * v[0] + v[1] * v[1] + v[2] * v[2] + v[3] * v[3];
    #pragma unroll
    for (int o = 16; o > 0; o >>= 1) s += __shfl_xor(s, o, 32);
    if (lane == 0) norms[wave] = s;
}

// ---- 16 edge dot-products per wave via V_WMMA_F32_16X16X4_F32 diagonal ----
// f32 A(16x4)/B(4x16) layouts are symmetric: lane L supplies elements
// [k0 + 2*(L>>4), +1] of row (L&15) for both operands -> two b64 loads/chunk,
// every row byte fetched exactly once.
__device__ __forceinline__ v8f edge_dots_wmma(const float* __restrict__ ca,
                                              const float* __restrict__ pa,
                                              int koff) {
    v8f acc = {};
    #pragma unroll
    for (int k0 = 0; k0 < D; k0 += 4) {
        v2f a = *(const v2f*)(ca + k0 + koff);
        v2f b = *(const v2f*)(pa + k0 + koff);
        acc = __builtin_amdgcn_wmma_f32_16x16x4_f32(false, a, false, b,
                                                    (short)0, acc, false, false);
    }
    return acc;
}

// diag(i,i): i<8 -> lane i, acc[i]; i>=8 -> lane i+16 (lanes 24..31), acc[i-8]
__device__ __forceinline__ float diag_pick(const v8f& acc, int r) {
    return r == 0 ? acc[0] : r == 1 ? acc[1] : r == 2 ? acc[2] : r == 3 ? acc[3]
         : r == 4 ? acc[4] : r == 5 ? acc[5] : r == 6 ? acc[6] : acc[7];
}

template <bool IS_POS>
__global__ void edge_kernel(const float* __restrict__ g, const float* __restrict__ norms,
                            const int* __restrict__ e0, const int* __restrict__ e1,
                            float* __restrict__ p_sum, unsigned* __restrict__ p_cnt,
                            unsigned* __restrict__ n_key, int E) {
    int wave = (blockIdx.x * blockDim.x + threadIdx.x) >> 5;
    int lane = threadIdx.x & 31;
    int base = wave * 16;
    if (base >= E) return;               // wave-uniform: EXEC stays all-ones

    int m = lane & 15;                   // edge slot within the wave's tile of 16
    int e = base + m;
    int ec = (e < E) ? e : (E - 1);      // clamp tail without lane divergence
    int ci = __builtin_nontemporal_load(e0 + ec);   // edge lists: single touch
    int pi = __builtin_nontemporal_load(e1 + ec);
    const float* ca = g + (size_t)ci * D;
    const float* pa = g + (size_t)pi * D;
    int koff = (lane >> 4) << 1;         // lanes 0-15: K pair {0,1}; 16-31: {2,3}

    v8f acc = edge_dots_wmma(ca, pa, koff);

    bool low  = (lane < 8);
    bool high = (lane >= 24);
    if ((low || high) && e < E) {
        int r = low ? lane : (lane - 24);
        float dot = diag_pick(acc, r);
        float sim = 2.0f * dot - norms[ci] - norms[pi];   // = -||a-b||^2
        if (IS_POS) {
            atomicAdd(&p_sum[ci], sim);
            atomicAdd(&p_cnt[ci], 1u);
        } else {
            atomicMax(&n_key[ci], f2key(sim));
        }
    }
}

// ---- hinge mean over all nodes ----
__global__ void finalize_kernel(const float* __restrict__ p_sum, const unsigned* __restrict__ p_cnt,
                                const unsigned* __restrict__ n_key, float* __restrict__ out, int n) {
    int i = blockIdx.x * blockDim.x + threadIdx.x;
    float term = 0.0f;
    if (i < n) {
        float cnt = (float)p_cnt[i];
        float p_d = p_sum[i] / fmaxf(cnt, 1.0f);
        unsigned k = n_key[i];
        float n_d = (k != 0u) ? key2f(k) : 0.0f;          // key==0 <=> no neg edge
        term = fmaxf(n_d - p_d + DELTA_V, 0.0f);
    }
    #pragma unroll
    for (int o = 16; o > 0; o >>= 1) term += __shfl_xor(term, o, 32);
    if ((threadIdx.x & 31) == 0) atomicAdd(out, term / (float)n);
}

extern "C" void kernel_launch(void* const* d_in, const int* in_sizes, int n_in,
                              void* d_out, int out_size, void* d_ws, size_t ws_size,
                              hipStream_t stream) {
    const float* emb   = (const float*)d_in[0];
    const int*   nodes = (const int*)d_in[1];
    const int*   pos   = (const int*)d_in[2];
    const int*   neg   = (const int*)d_in[3];
    int n_nodes = in_sizes[1];
    int E       = in_sizes[2] / 2;     // edges stored as (2, E) row-major

    // workspace layout
    float*    g     = (float*)d_ws;                          // n_nodes * 128 f32 (64 MB)
    float*    norms = g + (size_t)n_nodes * D;               // n_nodes f32
    float*    p_sum = norms + n_nodes;                       // n_nodes f32
    unsigned* p_cnt = (unsigned*)(p_sum + n_nodes);          // n_nodes u32
    unsigned* n_key = p_cnt + n_nodes;                       // n_nodes u32
    float*    out   = (float*)d_out;

    const int BLK = 256;                                     // 8 wave32 per block

    zero_kernel<<<(n_nodes + BLK - 1) / BLK, BLK, 0, stream>>>(p_sum, p_cnt, n_key, out, n_nodes);

    int gather_blocks = (n_nodes + 7) / 8;                   // 1 wave per row
    gather_norm_kernel<<<gather_blocks, BLK, 0, stream>>>(emb, nodes, g, norms, n_nodes);

    int waves_e = (E + 15) / 16;                             // 16 edges per wave
    int edge_blocks = (waves_e + 7) / 8;
    edge_kernel<true ><<<edge_blocks, BLK, 0, stream>>>(g, norms, pos, pos + E,
                                                        p_sum, p_cnt, n_key, E);
    edge_kernel<false><<<edge_blocks, BLK, 0, stream>>>(g, norms, neg, neg + E,
                                                        p_sum, p_cnt, n_key, E);

    finalize_kernel<<<(n_nodes + BLK - 1) / BLK, BLK, 0, stream>>>(p_sum, p_cnt, n_key, out, n_nodes);
}